// RejectionSampler_80573586473414
// MI455X (gfx1250) — compile-verified
//
#include <hip/hip_runtime.h>
#include <stdint.h>

#define PLACEHOLDER  (-1)
#define V_SIZE       32000
#define B_SIZE       128
#define S_LEN        8
#define TINY_F       1.17549435e-38f
#define NEG_HUGE_F   (-3.402823466e+38f)

#define THREADS      256
#define VEC4_PER_ROW (V_SIZE / 4)                       // 8000
#define FULL_TILES   (VEC4_PER_ROW / THREADS)           // 31
#define TAIL_V4      (VEC4_PER_ROW - FULL_TILES*THREADS) // 64

// ---------------------------------------------------------------------------
// CDNA5 async global -> LDS copy (gfx1250). vdst = per-lane LDS byte address,
// vaddr = per-lane 32-bit byte offset, saddr = 64-bit SGPR base (GVS mode).
// Tracked with ASYNCcnt; one counter increment per instruction per wave.
// ---------------------------------------------------------------------------
__device__ __forceinline__ unsigned lds_off(const void* p) {
    // generic shared pointer: high 32 bits = LDS aperture, low 32 bits = offset
    return (unsigned)(unsigned long long)(uintptr_t)p;
}

__device__ __forceinline__ void async_cp_b128(unsigned lds_byte, const float* base,
                                              unsigned byte_off) {
    asm volatile("global_load_async_to_lds_b128 %0, %1, %2"
                 :
                 : "v"(lds_byte), "v"(byte_off),
                   "s"((unsigned long long)(uintptr_t)base)
                 : "memory");
}

#define ASYNC_WAIT_LE_3() asm volatile("s_wait_asynccnt 0x3" ::: "memory")
#define ASYNC_WAIT_LE_1() asm volatile("s_wait_asynccnt 0x1" ::: "memory")
#define ASYNC_WAIT_LE_0() asm volatile("s_wait_asynccnt 0x0" ::: "memory")

// ---------------------------------------------------------------------------
// Fused per-row argmax kernel.
// HAS_SCORE=true : one block per (b, s<8) row. Computes
//                  argmax(target[b,s,:]) and argmax(max(t-d,TINY)/exp_q[b,:])
//                  (sum-normalization dropped: positive scalar, argmax-invariant)
// HAS_SCORE=false: one block per b, row s=8, target argmax only.
// ---------------------------------------------------------------------------
template <bool HAS_SCORE>
__global__ __launch_bounds__(THREADS)
void argmax_rows_kernel(const float* __restrict__ target,
                        const float* __restrict__ draftp,
                        const float* __restrict__ expq,
                        int* __restrict__ targmax,
                        int* __restrict__ recovered) {
    __shared__ float4 sT[2][THREADS];
    __shared__ float4 sD[2][THREADS];
    __shared__ float4 sQ[2][THREADS];
    __shared__ float  rvT[THREADS / 32];
    __shared__ int    riT[THREADS / 32];
    __shared__ float  rvS[THREADS / 32];
    __shared__ int    riS[THREADS / 32];

    const unsigned tid = threadIdx.x;
    int b, s, row_t, row_d = 0;
    const float* trow;
    const float* drow = nullptr;
    const float* qrow = nullptr;

    if (HAS_SCORE) {
        const int r = blockIdx.x;     // 0 .. B*S-1
        b = r >> 3;
        s = r & 7;
        row_t = b * (S_LEN + 1) + s;
        row_d = b * S_LEN + s;
        trow = target + (size_t)row_t * V_SIZE;
        drow = draftp + (size_t)row_d * V_SIZE;
        qrow = expq   + (size_t)b * V_SIZE;
    } else {
        b = blockIdx.x;
        row_t = b * (S_LEN + 1) + S_LEN;
        trow = target + (size_t)row_t * V_SIZE;
    }

    float bT = NEG_HUGE_F; int iT = 0x7fffffff;
    float bS = NEG_HUGE_F; int iS = 0x7fffffff;

    auto issue = [&](int t) {
        const unsigned par  = (unsigned)t & 1u;
        const unsigned goff = (unsigned)((t * THREADS + (int)tid) * 16);
        async_cp_b128(lds_off(&sT[par][tid]), trow, goff);
        if (HAS_SCORE) {
            async_cp_b128(lds_off(&sD[par][tid]), drow, goff);
            async_cp_b128(lds_off(&sQ[par][tid]), qrow, goff);
        }
    };

    auto scan4 = [&](int base, const float4& t4, const float4& d4, const float4& q4) {
        const float* tp = (const float*)&t4;
        const float* dp = (const float*)&d4;
        const float* qp = (const float*)&q4;
#pragma unroll
        for (int j = 0; j < 4; ++j) {
            const float tv = tp[j];
            if (tv > bT) { bT = tv; iT = base + j; }   // strict > keeps first index
            if (HAS_SCORE) {
                float a = tv - dp[j];
                a = (a < TINY_F) ? TINY_F : a;
                const float sc = a / qp[j];
                if (sc > bS) { bS = sc; iS = base + j; }
            }
        }
    };

    // ---- double-buffered async pipeline over 31 full tiles -----------------
    issue(0);
    for (int t = 0; t < FULL_TILES; ++t) {
        if (t + 1 < FULL_TILES) {
            issue(t + 1);
            if (HAS_SCORE) ASYNC_WAIT_LE_3();   // tile t (3 ops) complete
            else           ASYNC_WAIT_LE_1();   // tile t (1 op) complete
        } else {
            ASYNC_WAIT_LE_0();                  // drain
        }
        const unsigned par = (unsigned)t & 1u;
        float4 t4 = sT[par][tid];
        float4 d4 = {}, q4 = {};
        if (HAS_SCORE) { d4 = sD[par][tid]; q4 = sQ[par][tid]; }
        scan4((t * THREADS + (int)tid) * 4, t4, d4, q4);
    }

    // ---- ragged tail (64 vec4) via direct global b128 loads ----------------
    if (tid < TAIL_V4) {
        const int v4i = FULL_TILES * THREADS + (int)tid;
        float4 t4 = ((const float4*)trow)[v4i];
        float4 d4 = {}, q4 = {};
        if (HAS_SCORE) {
            d4 = ((const float4*)drow)[v4i];
            q4 = ((const float4*)qrow)[v4i];
        }
        scan4(v4i * 4, t4, d4, q4);
    }

    // ---- wave32 shuffle reduction (max value, min index on tie) ------------
#pragma unroll
    for (int off = 16; off > 0; off >>= 1) {
        float ov = __shfl_xor(bT, off, 32);
        int   oi = __shfl_xor(iT, off, 32);
        if (ov > bT || (ov == bT && oi < iT)) { bT = ov; iT = oi; }
        if (HAS_SCORE) {
            float sv = __shfl_xor(bS, off, 32);
            int   si = __shfl_xor(iS, off, 32);
            if (sv > bS || (sv == bS && si < iS)) { bS = sv; iS = si; }
        }
    }

    const int wave = (int)(tid >> 5);
    const int lane = (int)(tid & 31);
    if (lane == 0) {
        rvT[wave] = bT; riT[wave] = iT;
        if (HAS_SCORE) { rvS[wave] = bS; riS[wave] = iS; }
    }
    __syncthreads();
    if (tid == 0) {
        float v = rvT[0]; int i = riT[0];
#pragma unroll
        for (int w = 1; w < THREADS / 32; ++w)
            if (rvT[w] > v || (rvT[w] == v && riT[w] < i)) { v = rvT[w]; i = riT[w]; }
        targmax[row_t] = i;
        if (HAS_SCORE) {
            float v2 = rvS[0]; int i2 = riS[0];
#pragma unroll
            for (int w = 1; w < THREADS / 32; ++w)
                if (rvS[w] > v2 || (rvS[w] == v2 && riS[w] < i2)) { v2 = rvS[w]; i2 = riS[w]; }
            recovered[row_d] = i2;
        }
    }
}

// ---------------------------------------------------------------------------
// Finalize: one thread per batch row. Sequential accept scan with early exit,
// token gathers for the random-accept test, bonus/rejection handling.
// ---------------------------------------------------------------------------
__global__ void finalize_kernel(const int*   __restrict__ draft_tok,
                                const float* __restrict__ target,
                                const float* __restrict__ draftp,
                                const int*   __restrict__ bonus,
                                const float* __restrict__ uni,
                                const int*   __restrict__ is_greedy,
                                const int*   __restrict__ targmax,
                                const int*   __restrict__ recovered,
                                float*       __restrict__ out) {
    const int b = (int)(blockIdx.x * blockDim.x + threadIdx.x);
    if (b >= B_SIZE) return;

    float o[S_LEN + 1];
#pragma unroll
    for (int j = 0; j <= S_LEN; ++j) o[j] = -1.0f;

    const bool g = is_greedy[b] != 0;
    int  k = 0;
    bool rejected = false;

    for (int p = 0; p < S_LEN; ++p) {
        const int tok = draft_tok[b * S_LEN + p];
        if (tok == PLACEHOLDER) break;          // invalid: stop, no emit
        const int ta = targmax[b * (S_LEN + 1) + p];
        bool acc;
        if (g) {
            acc = (tok == ta);
        } else {
            const float tp = target[(size_t)(b * (S_LEN + 1) + p) * V_SIZE + tok];
            const float dp = draftp[(size_t)(b * S_LEN + p) * V_SIZE + tok];
            acc = (tp / dp) >= uni[b * S_LEN + p];
        }
        if (acc) {
            o[p] = (float)tok; ++k;
        } else {
            const int rep = g ? ta : recovered[b * S_LEN + p];
            o[p] = (float)rep; ++k;
            rejected = true;
            break;                              // rejected position emits, then stop
        }
    }
    o[k] = rejected ? -1.0f : (float)bonus[b];

#pragma unroll
    for (int j = 0; j <= S_LEN; ++j) out[b * (S_LEN + 1) + j] = o[j];
}

// ---------------------------------------------------------------------------
extern "C" void kernel_launch(void* const* d_in, const int* in_sizes, int n_in,
                              void* d_out, int out_size, void* d_ws, size_t ws_size,
                              hipStream_t stream) {
    (void)in_sizes; (void)n_in; (void)out_size; (void)ws_size;

    const int*   draft_tok = (const int*)  d_in[0];   // [B,S]   int
    const float* draft_p   = (const float*)d_in[1];   // [B,S,V] f32
    const float* target_p  = (const float*)d_in[2];   // [B,S+1,V] f32
    const int*   bonus     = (const int*)  d_in[3];   // [B]     int
    const float* uni       = (const float*)d_in[4];   // [B,S]   f32
    const float* expq      = (const float*)d_in[5];   // [B,V]   f32
    const int*   greedy    = (const int*)  d_in[6];   // [B]     bool->int

    float* out       = (float*)d_out;                 // [B,S+1]
    int*   targmax   = (int*)d_ws;                    // [B,S+1]
    int*   recovered = targmax + B_SIZE * (S_LEN + 1);// [B,S]

    argmax_rows_kernel<true ><<<B_SIZE * S_LEN, THREADS, 0, stream>>>(
        target_p, draft_p, expq, targmax, recovered);
    argmax_rows_kernel<false><<<B_SIZE,        THREADS, 0, stream>>>(
        target_p, nullptr, nullptr, targmax, recovered);
    finalize_kernel<<<1, B_SIZE, 0, stream>>>(
        draft_tok, target_p, draft_p, bonus, uni, greedy, targmax, recovered, out);
}